// SpectralConv_30717606101415
// MI455X (gfx1250) — compile-verified
//
#include <hip/hip_runtime.h>

// ---------------------------------------------------------------------------
// SpectralConv (FNO) on gfx1250 via V_WMMA_F32_16X16X4_F32.
// Pipeline: twiddles -> expand core -> 5 WMMA GEMM stages.
// All 31-mode dims padded to 32 with hard zeros so EXEC is all-1s at WMMA.
// ---------------------------------------------------------------------------

typedef __attribute__((ext_vector_type(2))) float v2f;
typedef __attribute__((ext_vector_type(8))) float v8f;

#define CCH 64     // channels
#define NPIX 128   // nx == ny
#define NBATCH 32
#define NMODE 32   // padded modes (31 real + 1 zero)

__device__ __forceinline__ v8f wmma4(v2f a, v2f b, v8f c) {
  // D = A(16x4) * B(4x16) + C(16x16), all f32, wave32.
  return __builtin_amdgcn_wmma_f32_16x16x4_f32(false, a, false, b, (short)0, c,
                                               false, false);
}

// --------------------------- twiddle matrices ------------------------------
// MxR/MxI: (32 x 128) row-major, row 31 zeroed.  iMxR/iMxI: (128 x 32), col31=0.
__global__ void k_build_mats(float* __restrict__ MxR, float* __restrict__ MxI,
                             float* __restrict__ iMxR, float* __restrict__ iMxI) {
  int t = blockIdx.x * blockDim.x + threadIdx.x;
  if (t >= NMODE * NPIX) return;
  int X = t >> 7;        // 0..31
  int xx = t & 127;      // 0..127
  float vR = 0.f, vI = 0.f;
  if (X < 31) {
    int f = (X < 16) ? X : (97 + X);            // {0..15, 113..127}
    int ph = (f * xx) & 127;                    // exact phase mod n
    float ang = (float)ph * 0.049087385212340519f; // 2*pi/128
    vR = __cosf(ang);
    vI = -__sinf(ang);
  }
  MxR[X * NPIX + xx] = vR;
  MxI[X * NPIX + xx] = vI;
  iMxR[xx * NMODE + X] = vR * (1.0f / 128.0f);   // cos/n
  iMxI[xx * NMODE + X] = -vI * (1.0f / 128.0f);  // +sin/n
}

// --------------------------- core expansion --------------------------------
// coreR/coreI layout: [X(32)][Y(32)][o(64)][c(64)], zeros at X==31 || Y==31.
__global__ void k_expand_core(const float* __restrict__ w,
                              float* __restrict__ coreR,
                              float* __restrict__ coreI) {
  size_t t = (size_t)blockIdx.x * blockDim.x + threadIdx.x;
  if (t >= (size_t)NMODE * NMODE * CCH * CCH) return;
  int c = (int)(t & 63);
  int o = (int)((t >> 6) & 63);
  int Y = (int)((t >> 12) & 31);
  int X = (int)(t >> 17);
  float vr = 0.f, vi = 0.f;
  if (X < 31 && Y < 31) {
    int ix = (X + 15) % 31;            // ifftshift over NCX
    int iy = (Y + 15) % 31;            // ifftshift over NCY
    int oo = (o + 32) & 63;            // ifftshift over C (out)
    int cc = (c + 32) & 63;            // ifftshift over C (in)
    int j1 = iy * 31 + ix;
    int j2 = 960 - j1;                 // flip over last two axes
    const float* wp = w + ((size_t)oo * 64 + cc) * 961;
    float r1 = (j1 >= 480) ? wp[j1 - 480] : 0.f;
    float r2 = (j2 >= 480) ? wp[j2 - 480] : 0.f;
    float i1 = (j1 >= 481) ? wp[j1] : 0.f;
    float i2 = (j2 >= 481) ? wp[j2] : 0.f;
    vr = r1 + r2;
    vi = i1 - i2;
  }
  coreR[t] = vr;
  coreI[t] = vi;
}

// --------------------- stage 1: truncated DFT over x -----------------------
// per (b,y): F1[X(32), c(64)] = Mx(32x128) @ xslice(128x64); 256 thr / 8 waves
__global__ void k_stage1(const float* __restrict__ x,
                         const float* __restrict__ MxR,
                         const float* __restrict__ MxI,
                         float* __restrict__ F1R, float* __restrict__ F1I) {
  __shared__ float lds[NPIX * CCH]; // 32 KB: x[b,:,y,:] as [xx][c]
  int bid = blockIdx.x;
  int b = bid >> 7, y = bid & 127;
  const float* xs = x + (size_t)b * (NPIX * NPIX * CCH) + (size_t)y * CCH;
  for (int l = threadIdx.x; l < NPIX * CCH; l += 256) {
    int xx = l >> 6, c = l & 63;
    lds[l] = xs[(size_t)xx * (NPIX * CCH) + c];
  }
  __syncthreads();
  int wave = threadIdx.x >> 5, lane = threadIdx.x & 31;
  int half = lane >> 4, r = lane & 15;
  int m0 = (wave >> 2) * 16, n0 = (wave & 3) * 16;
  v8f accR = {0, 0, 0, 0, 0, 0, 0, 0};
  v8f accI = {0, 0, 0, 0, 0, 0, 0, 0};
#pragma unroll 4
  for (int k0 = 0; k0 < NPIX; k0 += 4) {
    int ka = k0 + 2 * half;
    v2f aR, aI, bf;
    aR.x = MxR[(m0 + r) * NPIX + ka];     aR.y = MxR[(m0 + r) * NPIX + ka + 1];
    aI.x = MxI[(m0 + r) * NPIX + ka];     aI.y = MxI[(m0 + r) * NPIX + ka + 1];
    bf.x = lds[ka * CCH + n0 + r];        bf.y = lds[(ka + 1) * CCH + n0 + r];
    accR = wmma4(aR, bf, accR);
    accI = wmma4(aI, bf, accI);
  }
  size_t base = ((size_t)b * NMODE) * (NPIX * CCH) + (size_t)y * CCH;
#pragma unroll
  for (int i = 0; i < 8; ++i) {
    int X = m0 + 8 * half + i;
    size_t idx = base + (size_t)X * (NPIX * CCH) + (n0 + r);
    F1R[idx] = accR[i];
    F1I[idx] = accI[i];
  }
}

// --------------------- stage 2: truncated DFT over y -----------------------
// per (b,X): F2[Y(32), c] = My(cplx) @ F1slice(128x64 cplx); K split in two.
__global__ void k_stage2(const float* __restrict__ F1R,
                         const float* __restrict__ F1I,
                         const float* __restrict__ MR,
                         const float* __restrict__ MI,
                         float* __restrict__ F2R, float* __restrict__ F2I) {
  __shared__ float ldsR[64 * CCH]; // 16 KB
  __shared__ float ldsI[64 * CCH]; // 16 KB
  int bid = blockIdx.x;
  int b = bid >> 5, X = bid & 31;
  const float* sR = F1R + ((size_t)b * NMODE + X) * (NPIX * CCH);
  const float* sI = F1I + ((size_t)b * NMODE + X) * (NPIX * CCH);
  int wave = threadIdx.x >> 5, lane = threadIdx.x & 31;
  int half = lane >> 4, r = lane & 15;
  int m0 = (wave >> 2) * 16, n0 = (wave & 3) * 16;
  v8f accR = {0, 0, 0, 0, 0, 0, 0, 0};
  v8f accI = {0, 0, 0, 0, 0, 0, 0, 0};
  for (int kk = 0; kk < 2; ++kk) {
    __syncthreads();
    for (int l = threadIdx.x; l < 64 * CCH; l += 256) {
      ldsR[l] = sR[kk * 64 * CCH + l];
      ldsI[l] = sI[kk * 64 * CCH + l];
    }
    __syncthreads();
#pragma unroll 4
    for (int k0 = 0; k0 < 64; k0 += 4) {
      int ka = k0 + 2 * half;
      int kg = kk * 64 + ka;
      v2f aR, aI, bR, bI;
      aR.x = MR[(m0 + r) * NPIX + kg];  aR.y = MR[(m0 + r) * NPIX + kg + 1];
      aI.x = MI[(m0 + r) * NPIX + kg];  aI.y = MI[(m0 + r) * NPIX + kg + 1];
      v2f aIn = -aI;
      bR.x = ldsR[ka * CCH + n0 + r];   bR.y = ldsR[(ka + 1) * CCH + n0 + r];
      bI.x = ldsI[ka * CCH + n0 + r];   bI.y = ldsI[(ka + 1) * CCH + n0 + r];
      accR = wmma4(aR, bR, accR);
      accR = wmma4(aIn, bI, accR);  // Re -= MI*F1I
      accI = wmma4(aR, bI, accI);
      accI = wmma4(aI, bR, accI);
    }
  }
  size_t base = ((size_t)b * NMODE + X) * (NMODE * CCH);
#pragma unroll
  for (int i = 0; i < 8; ++i) {
    int Y = m0 + 8 * half + i;
    size_t idx = base + (size_t)Y * CCH + (n0 + r);
    F2R[idx] = accR[i];
    F2I[idx] = accI[i];
  }
}

// ------------------ stage 3: per-mode complex channel mix ------------------
// per mode (X,Y): out[o(64), b(32)] = core(64x64 cplx) @ F2[c][b] (cplx)
__global__ void k_stage3(const float* __restrict__ F2R,
                         const float* __restrict__ F2I,
                         const float* __restrict__ coreR,
                         const float* __restrict__ coreI,
                         float* __restrict__ Y3R, float* __restrict__ Y3I) {
  __shared__ float ldsR[CCH * NBATCH]; // [c][b], 8 KB
  __shared__ float ldsI[CCH * NBATCH]; // 8 KB
  int mode = blockIdx.x;               // X*32 + Y
  for (int l = threadIdx.x; l < CCH * NBATCH; l += 256) {
    int c = l >> 5, bb = l & 31;
    size_t gi = ((size_t)bb * (NMODE * NMODE) + mode) * CCH + c;
    ldsR[c * NBATCH + bb] = F2R[gi];
    ldsI[c * NBATCH + bb] = F2I[gi];
  }
  __syncthreads();
  int wave = threadIdx.x >> 5, lane = threadIdx.x & 31;
  int half = lane >> 4, r = lane & 15;
  int m0 = (wave >> 1) * 16;           // o tiles: 0..3
  int n0 = (wave & 1) * 16;            // b tiles: 0..1
  const float* aRp = coreR + (size_t)mode * (CCH * CCH);
  const float* aIp = coreI + (size_t)mode * (CCH * CCH);
  v8f accR = {0, 0, 0, 0, 0, 0, 0, 0};
  v8f accI = {0, 0, 0, 0, 0, 0, 0, 0};
#pragma unroll 4
  for (int k0 = 0; k0 < CCH; k0 += 4) {
    int ka = k0 + 2 * half;
    v2f aR, aI, bR, bI;
    aR.x = aRp[(m0 + r) * CCH + ka];  aR.y = aRp[(m0 + r) * CCH + ka + 1];
    aI.x = aIp[(m0 + r) * CCH + ka];  aI.y = aIp[(m0 + r) * CCH + ka + 1];
    v2f aIn = -aI;
    bR.x = ldsR[ka * NBATCH + n0 + r];  bR.y = ldsR[(ka + 1) * NBATCH + n0 + r];
    bI.x = ldsI[ka * NBATCH + n0 + r];  bI.y = ldsI[(ka + 1) * NBATCH + n0 + r];
    accR = wmma4(aR, bR, accR);
    accR = wmma4(aIn, bI, accR);
    accI = wmma4(aR, bI, accI);
    accI = wmma4(aI, bR, accI);
  }
#pragma unroll
  for (int i = 0; i < 8; ++i) {
    int o = m0 + 8 * half + i;
    int bb = n0 + r;
    size_t idx = ((size_t)bb * (NMODE * NMODE) + mode) * CCH + o;
    Y3R[idx] = accR[i];
    Y3I[idx] = accI[i];
  }
}

// ---------------------- stage 4: inverse DFT over x ------------------------
// per (b,Y): O4[x(128), o(64)] = iMx(128x32 cplx) @ Y3[X][o] (cplx); 1024 thr
__global__ void k_stage4(const float* __restrict__ Y3R,
                         const float* __restrict__ Y3I,
                         const float* __restrict__ iMR,
                         const float* __restrict__ iMI,
                         float* __restrict__ O4R, float* __restrict__ O4I) {
  __shared__ float ldsR[NMODE * CCH]; // 8 KB, [X][o]
  __shared__ float ldsI[NMODE * CCH]; // 8 KB
  int bid = blockIdx.x;
  int b = bid >> 5, Y = bid & 31;
  for (int l = threadIdx.x; l < NMODE * CCH; l += 1024) {
    int X = l >> 6, o = l & 63;
    size_t gi = (((size_t)b * NMODE + X) * NMODE + Y) * CCH + o;
    ldsR[l] = Y3R[gi];
    ldsI[l] = Y3I[gi];
  }
  __syncthreads();
  int wave = threadIdx.x >> 5, lane = threadIdx.x & 31;
  int half = lane >> 4, r = lane & 15;
  int m0 = (wave >> 2) * 16;   // x tiles 0..7
  int n0 = (wave & 3) * 16;    // o tiles 0..3
  v8f accR = {0, 0, 0, 0, 0, 0, 0, 0};
  v8f accI = {0, 0, 0, 0, 0, 0, 0, 0};
#pragma unroll
  for (int k0 = 0; k0 < NMODE; k0 += 4) {
    int ka = k0 + 2 * half;
    v2f aR, aI, bR, bI;
    aR.x = iMR[(m0 + r) * NMODE + ka];  aR.y = iMR[(m0 + r) * NMODE + ka + 1];
    aI.x = iMI[(m0 + r) * NMODE + ka];  aI.y = iMI[(m0 + r) * NMODE + ka + 1];
    v2f aIn = -aI;
    bR.x = ldsR[ka * CCH + n0 + r];  bR.y = ldsR[(ka + 1) * CCH + n0 + r];
    bI.x = ldsI[ka * CCH + n0 + r];  bI.y = ldsI[(ka + 1) * CCH + n0 + r];
    accR = wmma4(aR, bR, accR);
    accR = wmma4(aIn, bI, accR);
    accI = wmma4(aR, bI, accI);
    accI = wmma4(aI, bR, accI);
  }
#pragma unroll
  for (int i = 0; i < 8; ++i) {
    int xr = m0 + 8 * half + i;
    size_t idx = (((size_t)b * NPIX + xr) * NMODE + Y) * CCH + (n0 + r);
    O4R[idx] = accR[i];
    O4I[idx] = accI[i];
  }
}

// ------------------ stage 5: inverse DFT over y, real out ------------------
// per (b,x): out[y(128), o(64)] = Re(iMy @ O4slice) ; 1024 thr
__global__ void k_stage5(const float* __restrict__ O4R,
                         const float* __restrict__ O4I,
                         const float* __restrict__ iMR,
                         const float* __restrict__ iMI,
                         float* __restrict__ out) {
  __shared__ float ldsR[NMODE * CCH]; // 8 KB, [Y][o]
  __shared__ float ldsI[NMODE * CCH]; // 8 KB
  int bid = blockIdx.x;
  int b = bid >> 7, xr = bid & 127;
  size_t base = ((size_t)b * NPIX + xr) * (NMODE * CCH);
  for (int l = threadIdx.x; l < NMODE * CCH; l += 1024) {
    ldsR[l] = O4R[base + l];
    ldsI[l] = O4I[base + l];
  }
  __syncthreads();
  int wave = threadIdx.x >> 5, lane = threadIdx.x & 31;
  int half = lane >> 4, r = lane & 15;
  int m0 = (wave >> 2) * 16;   // y tiles 0..7
  int n0 = (wave & 3) * 16;    // o tiles 0..3
  v8f accR = {0, 0, 0, 0, 0, 0, 0, 0};
#pragma unroll
  for (int k0 = 0; k0 < NMODE; k0 += 4) {
    int ka = k0 + 2 * half;
    v2f aR, aI, bR, bI;
    aR.x = iMR[(m0 + r) * NMODE + ka];  aR.y = iMR[(m0 + r) * NMODE + ka + 1];
    aI.x = iMI[(m0 + r) * NMODE + ka];  aI.y = iMI[(m0 + r) * NMODE + ka + 1];
    v2f aIn = -aI;
    bR.x = ldsR[ka * CCH + n0 + r];  bR.y = ldsR[(ka + 1) * CCH + n0 + r];
    bI.x = ldsI[ka * CCH + n0 + r];  bI.y = ldsI[(ka + 1) * CCH + n0 + r];
    accR = wmma4(aR, bR, accR);
    accR = wmma4(aIn, bI, accR);  // Re = RR - II
  }
  size_t ob = ((size_t)b * NPIX + xr) * (NPIX * CCH);
#pragma unroll
  for (int i = 0; i < 8; ++i) {
    int yr = m0 + 8 * half + i;
    out[ob + (size_t)yr * CCH + (n0 + r)] = accR[i];
  }
}

// ----------------------------- launcher ------------------------------------
extern "C" void kernel_launch(void* const* d_in, const int* in_sizes, int n_in,
                              void* d_out, int out_size, void* d_ws,
                              size_t ws_size, hipStream_t stream) {
  (void)in_sizes; (void)n_in; (void)out_size; (void)ws_size;
  const float* x = (const float*)d_in[0];   // (32,128,128,64) f32
  const float* w = (const float*)d_in[1];   // (961*64*64,) f32
  float* out = (float*)d_out;               // (32,128,128,64) f32
  float* ws = (float*)d_ws;

  // workspace layout (float offsets); O4 aliases F1 (dead after stage 2)
  const size_t oMxR  = 0;
  const size_t oMxI  = oMxR  + (size_t)NMODE * NPIX;          // 4096
  const size_t oiMxR = oMxI  + (size_t)NMODE * NPIX;
  const size_t oiMxI = oiMxR + (size_t)NPIX * NMODE;
  const size_t oCoR  = oiMxI + (size_t)NPIX * NMODE;          // 16384
  const size_t oCoI  = oCoR  + (size_t)NMODE * NMODE * CCH * CCH;
  const size_t oF1R  = oCoI  + (size_t)NMODE * NMODE * CCH * CCH;
  const size_t oF1I  = oF1R  + (size_t)NBATCH * NMODE * NPIX * CCH;
  const size_t oF2R  = oF1I  + (size_t)NBATCH * NMODE * NPIX * CCH;
  const size_t oF2I  = oF2R  + (size_t)NBATCH * NMODE * NMODE * CCH;
  const size_t oY3R  = oF2I  + (size_t)NBATCH * NMODE * NMODE * CCH;
  const size_t oY3I  = oY3R  + (size_t)NBATCH * NMODE * NMODE * CCH;
  const size_t oO4R  = oF1R;  // alias
  const size_t oO4I  = oF1I;  // alias

  float *MxR = ws + oMxR, *MxI = ws + oMxI;
  float *iMxR = ws + oiMxR, *iMxI = ws + oiMxI;
  float *coreR = ws + oCoR, *coreI = ws + oCoI;
  float *F1R = ws + oF1R, *F1I = ws + oF1I;
  float *F2R = ws + oF2R, *F2I = ws + oF2I;
  float *Y3R = ws + oY3R, *Y3I = ws + oY3I;
  float *O4R = ws + oO4R, *O4I = ws + oO4I;

  k_build_mats<<<(NMODE * NPIX + 255) / 256, 256, 0, stream>>>(MxR, MxI, iMxR, iMxI);
  k_expand_core<<<(NMODE * NMODE * CCH * CCH + 255) / 256, 256, 0, stream>>>(w, coreR, coreI);
  k_stage1<<<NBATCH * NPIX, 256, 0, stream>>>(x, MxR, MxI, F1R, F1I);
  k_stage2<<<NBATCH * NMODE, 256, 0, stream>>>(F1R, F1I, MxR, MxI, F2R, F2I);
  k_stage3<<<NMODE * NMODE, 256, 0, stream>>>(F2R, F2I, coreR, coreI, Y3R, Y3I);
  k_stage4<<<NBATCH * NMODE, 1024, 0, stream>>>(Y3R, Y3I, iMxR, iMxI, O4R, O4I);
  k_stage5<<<NBATCH * NPIX, 1024, 0, stream>>>(O4R, O4I, iMxR, iMxI, out);
}